// CEMA_23837068493125
// MI455X (gfx1250) — compile-verified
//
#include <hip/hip_runtime.h>
#include <math.h>

// EMA over batch-mean with broadcast:  out[b,s,d] = e[s,d],
//   e[s] = a*e[s-1] + (1-a)*mean_b x[b,s,d],  a = 0.99
//
// B=4, S=8192, D=1024, f32.  Memory-bound: min traffic 256MB -> ~11us @23.3TB/s.
// Scheme: mean (x->m, m lives in d_out b=0 slice & stays L2-resident),
// chunk-end weighted sums, per-column carry scan over 64 chunks, then the
// per-chunk scan expressed as lower-triangular 16x16 matmuls on the WMMA pipe
// (V_WMMA_F32_16X16X4_F32, exact fp32).

#define ALPHA_F 0.99f
#define CA_F    0.01f
#define B_      4
#define S_      8192
#define D_      1024
#define L_      128
#define NCHUNK  64      // S_/L_

typedef __attribute__((ext_vector_type(2))) float v2f;
typedef __attribute__((ext_vector_type(8))) float v8f;

// ---------------- Pass 1a: batch mean, m[s,d] = 0.25 * sum_b x[b,s,d] -------
__global__ void ema_mean_kernel(const float* __restrict__ x, float* __restrict__ m) {
    const size_t n = (size_t)S_ * D_;
    size_t i = (size_t)blockIdx.x * blockDim.x + threadIdx.x;
    float acc = x[i] + x[n + i] + x[2 * n + i] + x[3 * n + i];
    m[i] = 0.25f * acc;
}

// ---- Pass 1b: localEnd[c,d] = CA * sum_{k=0}^{L-1} a^{L-1-k} m[c*L+k, d] ----
__global__ void ema_chunk_end_kernel(const float* __restrict__ m, float* __restrict__ lend) {
    const int chunk = blockIdx.x;
    const int col   = blockIdx.y * blockDim.x + threadIdx.x;
    const float* mp = m + (size_t)chunk * L_ * D_ + col;
    float w = 1.0f, acc = 0.0f;
    for (int k = L_ - 1; k >= 0; --k) {      // weight a^{L-1-k}
        acc += w * mp[(size_t)k * D_];
        w *= ALPHA_F;
    }
    lend[chunk * D_ + col] = CA_F * acc;
}

// ---- Pass 2: carry[c,d] = EMA state just before chunk c (serial over 64) ----
__global__ void ema_carry_kernel(const float* __restrict__ lend,
                                 const float* __restrict__ ema0,
                                 float* __restrict__ carry, float aL) {
    const int d = threadIdx.x;               // blockDim.x == D_
    float e = ema0[d];
    for (int c = 0; c < NCHUNK; ++c) {
        carry[c * D_ + d] = e;
        e = aL * e + lend[c * D_ + d];
    }
}

// ---- Pass 3: per-chunk scan via WMMA triangular matmul, broadcast store ----
// e[sbase+i, n] = sum_{j<=i} CA*a^{i-j} * m[sbase+j, n]  +  a^{i+1} * carry[n]
// One wave per 16-column tile; 8 sequential 16-step segments per 128-chunk.
__global__ void __launch_bounds__(256)
ema_scan_wmma_kernel(const float* __restrict__ m, const float* __restrict__ carry,
                     float* __restrict__ out) {
    const int chunk = blockIdx.x;            // 0..63
    const int cg    = blockIdx.y;            // 0..7  (128 columns each)
    const int wave  = threadIdx.x >> 5;      // 0..7
    const int lane  = threadIdx.x & 31;
    const int half  = lane >> 4;             // 0: lanes 0-15, 1: lanes 16-31
    const int lrow  = lane & 15;

    const int ncol  = cg * 128 + wave * 16 + lrow;   // this lane's column

    // alpha powers (registers)
    const float a1 = ALPHA_F;
    const float a2 = a1 * a1, a4 = a2 * a2, a8 = a4 * a4;
    const float a9 = a8 * a1;
    const float inva  = 1.0f / ALPHA_F;
    const float inva2 = inva * inva, inva4 = inva2 * inva2;

    // rA = a^lrow  (exponentiation by bits, no indexed array -> no scratch)
    float rA = 1.0f;
    if (lrow & 1) rA *= a1;
    if (lrow & 2) rA *= a2;
    if (lrow & 4) rA *= a4;
    if (lrow & 8) rA *= a8;
    const float tRow  = CA_F * rA;           // CA * a^row
    const int   khalf = half * 2;            // A/B K-offset for this lane half
    const float invh  = half ? inva2 : 1.0f; // (1/a)^khalf

    float carryv = carry[chunk * D_ + ncol]; // e just before this chunk

    const int s0 = chunk * L_;
    for (int seg = 0; seg < 8; ++seg) {
        const int sbase = s0 + seg * 16;

        // C init: C[i][n] = a^{i+1} * carry[n];  lane holds rows half*8 .. +7
        v8f c;
        float w = carryv * (half ? a9 : a1);
        #pragma unroll
        for (int v = 0; v < 8; ++v) { c[v] = w; w *= a1; }

        // D += T(16x16, lower-tri) * M(16x16) as 4 chained K=4 WMMAs
        float invq = invh;                   // (1/a)^{4q + khalf}
        #pragma unroll
        for (int q = 0; q < 4; ++q) {
            const int j0 = 4 * q + khalf;    // global K of A.x / B.x
            v2f A, Bv;
            // T[i][j] = (i>=j) ? CA * a^i * (1/a)^j : 0
            A.x = (lrow >= j0)     ? tRow * invq        : 0.0f;
            A.y = (lrow >= j0 + 1) ? tRow * invq * inva : 0.0f;
            const size_t mb = (size_t)(sbase + j0) * D_ + ncol;
            Bv.x = m[mb];                    // row j0   of M tile
            Bv.y = m[mb + D_];               // row j0+1 of M tile
            c = __builtin_amdgcn_wmma_f32_16x16x4_f32(
                    false, A, false, Bv, (short)0, c, false, false);
            invq *= inva4;
        }

        // Store: lane's col ncol, rows half*8+v; broadcast to all 4 batches.
        const int rowbase = half * 8;
        const size_t bs = (size_t)S_ * D_;
        #pragma unroll
        for (int v = 0; v < 8; ++v) {
            const size_t o = (size_t)(sbase + rowbase + v) * D_ + ncol;
            float val = c[v];
            out[o]          = val;           // b=0 (overwrites consumed m rows)
            out[o + bs]     = val;
            out[o + 2 * bs] = val;
            out[o + 3 * bs] = val;
        }

        // Next segment carry = row 15 of this column = lane (lrow+16), c[7]
        carryv = __shfl(c[7], lrow + 16, 32);
    }
}

extern "C" void kernel_launch(void* const* d_in, const int* in_sizes, int n_in,
                              void* d_out, int out_size, void* d_ws, size_t ws_size,
                              hipStream_t stream) {
    (void)in_sizes; (void)n_in; (void)out_size; (void)ws_size;
    const float* x    = (const float*)d_in[0];   // [B,S,D] f32
    const float* ema0 = (const float*)d_in[1];   // [1,D]   f32
    float* out = (float*)d_out;                  // [B,S,D] f32

    float* m     = out;                          // b=0 slice doubles as m scratch
    float* lend  = (float*)d_ws;                 // [NCHUNK, D]
    float* carry = (float*)d_ws + NCHUNK * D_;   // [NCHUNK, D]

    const float aL = (float)pow((double)0.99, (double)L_);  // a^128 (host, double)

    // 1a: mean, x -> m (128MB read, 32MB write)
    ema_mean_kernel<<<(S_ * D_) / 256, 256, 0, stream>>>(x, m);
    // 1b: chunk-end weighted sums (m is L2-resident)
    ema_chunk_end_kernel<<<dim3(NCHUNK, D_ / 256), 256, 0, stream>>>(m, lend);
    // 2: per-column carry scan over 64 chunks
    ema_carry_kernel<<<1, D_, 0, stream>>>(lend, ema0, carry, aL);
    // 3: WMMA segmented scan + broadcast store (32MB read from L2, 128MB write)
    ema_scan_wmma_kernel<<<dim3(NCHUNK, D_ / 128), 256, 0, stream>>>(m, carry, out);
}